// HCMaps_36756330119661
// MI455X (gfx1250) — compile-verified
//
#include <hip/hip_runtime.h>
#include <cmath>

typedef __attribute__((ext_vector_type(2))) float v2f;
typedef __attribute__((ext_vector_type(8))) float v8f;

#define DIM          2048
#define TH_STRIDE    2052          // pad: dword bank = 4*m + 2h + k0 -> conflict-free
#define ROWS_TOTAL   32768         // B*T*N = 2*4096*4
#define HC_EPS       1e-6f

__launch_bounds__(256, 2)
__global__ void hcmaps_kernel(
    const float* __restrict__ x,          // [32768, 2048]
    const float* __restrict__ w,          // [2048] rms weight
    const float* __restrict__ alpha_pre_p,
    const float* __restrict__ alpha_post_p,
    const float* __restrict__ alpha_res_p,
    const float* __restrict__ theta_pre,  // [2048]
    const float* __restrict__ theta_post, // [2048]
    const float* __restrict__ theta_res,  // [4, 2048]
    const float* __restrict__ b_pre,      // [4]
    const float* __restrict__ b_post,     // [4]
    const float* __restrict__ b_res,      // [4,4]
    float* __restrict__ out)              // Hpre(32768) | Hpost(32768) | Hres(131072)
{
    __shared__ float sTheta[6 * TH_STRIDE];

    const int tid = threadIdx.x;

    // Phase 0: stage thetaW[m][c] = theta_m[c] * w[c] into LDS (6 rows).
    for (int i = tid; i < 6 * DIM; i += 256) {
        const int m = i >> 11;           // 0..5
        const int c = i & (DIM - 1);
        const float* src = (m == 0) ? theta_pre
                         : (m == 1) ? theta_post
                                    : (theta_res + (size_t)(m - 2) * DIM);
        sTheta[m * TH_STRIDE + c] = src[c] * w[c];
    }
    __syncthreads();

    // Phase 1: one wave owns a 16-row tile, full K=2048.
    // A (16x4) = thetaW slice, B (4x16) = x columns for the 16 rows.
    // Per ISA 32-bit operand layout: lane l -> (idx = l&15, half h = l>>4),
    // per K-step of 4 a lane supplies cols {k0+2h, k0+2h+1}.
    const int wave = tid >> 5;
    const int lane = tid & 31;
    const int r    = lane & 15;          // tile row (== WMMA N column, == A's M row)
    const int h    = lane >> 4;          // half
    const int rowBase = (blockIdx.x * 8 + wave) * 16;

    const float* xrow = x + (size_t)(rowBase + r) * DIM + 2 * h;
    const float* trow = sTheta + (r % 6) * TH_STRIDE + 2 * h;   // m>=6 dup row (D rows 6..15 unused)

    v8f acc = {0.f, 0.f, 0.f, 0.f, 0.f, 0.f, 0.f, 0.f};
    float ss = 0.f;

    // unroll 8: 8-deep global_load_b64 clause per body -> ~2KB in flight per wave
    #pragma unroll 8
    for (int k0 = 0; k0 < DIM; k0 += 4) {
        v2f a = *(const v2f*)(trow + k0);   // ds_load (pairs merge to ds_load_2addr_b64)
        v2f b = *(const v2f*)(xrow + k0);   // global_load_b64
        ss = fmaf(b.x, b.x, ss);
        ss = fmaf(b.y, b.y, ss);
        // D[m][n] += sum_k A[m][k] * B[k][n]   (fp32, exact-class numerics)
        acc = __builtin_amdgcn_wmma_f32_16x16x4_f32(
                  /*neg_a=*/false, a, /*neg_b=*/false, b,
                  /*c_mod=*/(short)0, acc, /*reuse_a=*/false, /*reuse_b=*/false);
    }

    // Sum-of-squares: lane l covered cols {2h,2h+1} mod 4 -> combine halves.
    ss += __shfl_xor(ss, 16, 32);

    // Phase 2: lane l<16 holds D[m][l] in acc[m] for m=0..7 -> all 6 results for its row.
    if (h == 0) {
        const int   R   = rowBase + r;
        const int   n   = R & 3;
        const int   bt  = R >> 2;
        const float rms = rsqrtf(ss * (1.0f / DIM) + HC_EPS);

        const float a_pre  = alpha_pre_p[0];
        const float a_post = alpha_post_p[0];
        const float a_res  = alpha_res_p[0];

        out[R]              = fmaf(a_pre,  tanhf(acc[0] * rms), b_pre[n]);
        out[ROWS_TOTAL + R] = fmaf(a_post, tanhf(acc[1] * rms), b_post[n]);
        #pragma unroll
        for (int i = 0; i < 4; ++i) {
            out[2 * ROWS_TOTAL + bt * 16 + i * 4 + n] =
                fmaf(a_res, tanhf(acc[2 + i] * rms), b_res[i * 4 + n]);
        }
    }
}

extern "C" void kernel_launch(void* const* d_in, const int* in_sizes, int n_in,
                              void* d_out, int out_size, void* d_ws, size_t ws_size,
                              hipStream_t stream) {
    // setup_inputs order:
    // 0 x_stream, 1 rms_weight, 2 alpha_pre, 3 alpha_post, 4 alpha_res,
    // 5 theta_pre, 6 theta_post, 7 theta_res, 8 b_pre, 9 b_post, 10 b_res
    hcmaps_kernel<<<dim3(ROWS_TOTAL / 128), dim3(256), 0, stream>>>(
        (const float*)d_in[0], (const float*)d_in[1],
        (const float*)d_in[2], (const float*)d_in[3], (const float*)d_in[4],
        (const float*)d_in[5], (const float*)d_in[6], (const float*)d_in[7],
        (const float*)d_in[8], (const float*)d_in[9], (const float*)d_in[10],
        (float*)d_out);
}